// ElmanNoDelta_86466281603869
// MI455X (gfx1250) — compile-verified
//
#include <hip/hip_runtime.h>
#include <cmath>

// Elman RNN (no delta), MI455X gfx1250, wave32 + V_WMMA_F32_16X16X4_F32.
// T=512, B=64, D=1024, alpha=0.88.

#define ALPHA     0.88f
#define T_STEPS   512
#define BDIM      64
#define DDIM      1024

typedef __attribute__((ext_vector_type(2))) float v2f;
typedef __attribute__((ext_vector_type(8))) float v8f;

// D(16x16,f32) = A(16x4,f32) * B(4x16,f32) + C   -- fp32 XDL path on CDNA5
__device__ __forceinline__ v8f wmma_f32(v2f a, v2f b, v8f c) {
    return __builtin_amdgcn_wmma_f32_16x16x4_f32(
        /*neg_a=*/false, a, /*neg_b=*/false, b,
        /*c_mod=*/(short)0, c, /*reuse_a=*/false, /*reuse_b=*/false);
}

// ---------------------------------------------------------------------------
// Phase 1: xw[m, n] = sum_k x[m, k] * Wx[n, k] + bias[n]
// M = T*B = 32768, N = D = 1024, K = D = 1024. Result written to d_out.
// Wave job = 2x2 block of 16x16 tiles (32 rows x 32 cols), 8 waves per WG.
// ---------------------------------------------------------------------------
__global__ __launch_bounds__(256)
void xw_gemm_kernel(const float* __restrict__ x, const float* __restrict__ Wx,
                    const float* __restrict__ bias, float* __restrict__ out)
{
    const int lane = threadIdx.x & 31;
    const int wave = threadIdx.x >> 5;
    const int job  = blockIdx.x * 8 + wave;     // 32768 wave-jobs total
    const int jn   = job & 31;                  // 32 n-jobs (32 cols each)
    const int jm   = job >> 5;                  // 1024 m-jobs (32 rows each)
    const int mBase = jm * 32;
    const int nBase = jn * 32;

    const int nl   = lane & 15;
    const int koff = (lane < 16) ? 0 : 2;       // A/B frag K sub-offset per half

    v8f acc00 = 0.0f, acc01 = 0.0f, acc10 = 0.0f, acc11 = 0.0f;

    const float* A0 = x  + (size_t)(mBase + nl)      * DDIM + koff;
    const float* A1 = x  + (size_t)(mBase + 16 + nl) * DDIM + koff;
    const float* B0 = Wx + (size_t)(nBase + nl)      * DDIM + koff;
    const float* B1 = Wx + (size_t)(nBase + 16 + nl) * DDIM + koff;

    #pragma unroll 4
    for (int k = 0; k < DDIM; k += 4) {
        v2f a0 = *(const v2f*)(A0 + k);
        v2f a1 = *(const v2f*)(A1 + k);
        v2f b0 = *(const v2f*)(B0 + k);
        v2f b1 = *(const v2f*)(B1 + k);
        acc00 = wmma_f32(a0, b0, acc00);
        acc01 = wmma_f32(a0, b1, acc01);
        acc10 = wmma_f32(a1, b0, acc10);
        acc11 = wmma_f32(a1, b1, acc11);
    }

    // Store with bias. C/D layout: row = v + (lane>=16 ? 8 : 0), col = lane%16.
    const int rsub = (lane < 16) ? 0 : 8;
    #pragma unroll
    for (int nt = 0; nt < 2; ++nt) {
        const int   col = nBase + nt * 16 + nl;
        const float bv  = bias[col];
        #pragma unroll
        for (int mt = 0; mt < 2; ++mt) {
            const int rbase = mBase + mt * 16 + rsub;
            v8f acc = (mt == 0) ? (nt == 0 ? acc00 : acc01)
                                : (nt == 0 ? acc10 : acc11);
            #pragma unroll
            for (int v = 0; v < 8; ++v)
                out[(size_t)(rbase + v) * DDIM + col] = acc[v] + bv;
        }
    }
}

// ---------------------------------------------------------------------------
// Phase 2: persistent recurrent kernel with grid-wide barrier per timestep.
//   cand = tanh(h @ R^T + xw_t);  h = a*h + (1-a)*cand;  hs[t] = h
// Grid: 64 WGs (one 16-wide n-slice each) x 4 waves (one 16-row b-tile each).
// The WG's R slice (16 rows x 1024 k = 64 KB) is staged in LDS once and
// reused for all 512 steps via conflict-free ds_load_b64 B-fragment reads.
// xw_t is read from d_out[t] and overwritten in place with h (same element).
// ---------------------------------------------------------------------------
__global__ __launch_bounds__(128)
void rnn_steps_kernel(const float* __restrict__ R, float* out,
                      float* hbuf0, float* hbuf1, unsigned* bar)
{
    // LDS image of B = R-slice, as K-pairs: Rlds[kp*16 + nl] = {R[n0+nl][2kp],
    // R[n0+nl][2kp+1]}. Fragment read: lanes 0-15 -> banks 0-31, lanes 16-31
    // -> banks 32-63 (conflict-free).
    __shared__ v2f Rlds[16 * (DDIM / 2)];       // 64 KB

    const int lane = threadIdx.x & 31;
    const int wave = threadIdx.x >> 5;
    const int n0   = blockIdx.x * 16;           // this WG's output columns
    const int b0   = wave * 16;                 // this wave's output rows
    const int nl   = lane & 15;
    const int koff = (lane < 16) ? 0 : 2;
    const int klo  = (lane < 16) ? 0 : 1;       // K-pair sub-offset per half
    const int rsub = (lane < 16) ? 0 : 8;
    const int col  = n0 + nl;

    // Cooperative one-time fill of the R slice (64 KB, amortized over 512 steps).
    for (int i = threadIdx.x; i < 16 * (DDIM / 2); i += 128) {
        const int rn = i & 15;                  // row within slice
        const int kp = i >> 4;                  // K-pair index
        Rlds[i] = *(const v2f*)(R + (size_t)(n0 + rn) * DDIM + 2 * kp);
    }
    __syncthreads();

    const v2f* Bp = Rlds + klo * 16 + nl;       // per-lane B-fragment base

    float* hc = hbuf0;   // current h  (pre-initialized to h0)
    float* hn = hbuf1;   // next h

    for (int t = 0; t < T_STEPS; ++t) {
        // Two interleaved accumulator chains over even/odd K quads.
        v8f acc0 = 0.0f, acc1 = 0.0f;
        const float* Arow = hc + (size_t)(b0 + nl) * DDIM + koff;
        #pragma unroll 4
        for (int k = 0; k < DDIM; k += 8) {
            v2f a0 = *(const v2f*)(Arow + k);
            v2f a1 = *(const v2f*)(Arow + k + 4);
            v2f bv0 = Bp[(k >> 1) * 16];
            v2f bv1 = Bp[(k >> 1) * 16 + 32];
            acc0 = wmma_f32(a0, bv0, acc0);
            acc1 = wmma_f32(a1, bv1, acc1);
        }
        v8f acc = acc0 + acc1;

        float* outT = out + (size_t)t * (BDIM * DDIM);
        #pragma unroll
        for (int v = 0; v < 8; ++v) {
            const int row = b0 + rsub + v;
            const float xw   = outT[(size_t)row * DDIM + col];
            const float hold = hc  [(size_t)row * DDIM + col];
            const float cand = tanhf(acc[v] + xw);
            const float hnew = ALPHA * hold + (1.0f - ALPHA) * cand;
            outT[(size_t)row * DDIM + col] = hnew;   // hs[t]
            hn  [(size_t)row * DDIM + col] = hnew;   // input to step t+1
        }

        if (t != T_STEPS - 1) {
            // grid-wide sense-reversal barrier: bar[0]=arrive count, bar[1]=generation
            __threadfence();
            __syncthreads();
            if (threadIdx.x == 0) {
                const unsigned gen =
                    __hip_atomic_load(&bar[1], __ATOMIC_ACQUIRE, __HIP_MEMORY_SCOPE_AGENT);
                const unsigned arrived =
                    __hip_atomic_fetch_add(&bar[0], 1u, __ATOMIC_ACQ_REL, __HIP_MEMORY_SCOPE_AGENT);
                if (arrived == gridDim.x - 1) {
                    __hip_atomic_store(&bar[0], 0u, __ATOMIC_RELAXED, __HIP_MEMORY_SCOPE_AGENT);
                    __hip_atomic_fetch_add(&bar[1], 1u, __ATOMIC_RELEASE, __HIP_MEMORY_SCOPE_AGENT);
                } else {
                    while (__hip_atomic_load(&bar[1], __ATOMIC_ACQUIRE,
                                             __HIP_MEMORY_SCOPE_AGENT) == gen) {
                        __builtin_amdgcn_s_sleep(1);
                    }
                }
            }
            __syncthreads();
        }

        float* tmp = hc; hc = hn; hn = tmp;
    }
}

// ---------------------------------------------------------------------------
extern "C" void kernel_launch(void* const* d_in, const int* in_sizes, int n_in,
                              void* d_out, int out_size, void* d_ws, size_t ws_size,
                              hipStream_t stream)
{
    const float* x    = (const float*)d_in[0];   // [T, B, D]
    const float* h0   = (const float*)d_in[1];   // [B, D]
    const float* Wx   = (const float*)d_in[2];   // [D, D]
    const float* R    = (const float*)d_in[3];   // [D, D]
    const float* bias = (const float*)d_in[4];   // [D]
    float* out = (float*)d_out;                  // [T, B, D]

    // Workspace: two h ping-pong buffers (256 KB each) + barrier words.
    float*    hbuf0 = (float*)d_ws;
    float*    hbuf1 = hbuf0 + (BDIM * DDIM);
    unsigned* bar   = (unsigned*)(hbuf1 + (BDIM * DDIM));

    // Reset barrier state and seed h with h0 (graph-capture-safe async ops).
    hipMemsetAsync((void*)bar, 0, 256, stream);
    hipMemcpyAsync((void*)hbuf0, (const void*)h0, (size_t)BDIM * DDIM * sizeof(float),
                   hipMemcpyDeviceToDevice, stream);

    // Phase 1: xw = x @ Wx^T + bias  -> d_out (consumed in place by phase 2)
    xw_gemm_kernel<<<dim3(4096), dim3(256), 0, stream>>>(x, Wx, bias, out);

    // Phase 2: 512 recurrent steps, persistent grid with per-step barrier.
    rnn_steps_kernel<<<dim3(64), dim3(128), 0, stream>>>(R, out, hbuf0, hbuf1, bar);
}